// FlowmatchingActionHead_51917564674058
// MI455X (gfx1250) — compile-verified
//
#include <hip/hip_runtime.h>
#include <hip/hip_bf16.h>
#include <math.h>

// ---- problem dims (compile-time constants from the reference) ----
#define BATCH   32
#define TSEQ    32
#define ADIM    16
#define HIDDEN  1536
#define NEMB    32

typedef __attribute__((ext_vector_type(2))) float v2f;
typedef __attribute__((ext_vector_type(8))) float v8f;

// -----------------------------------------------------------------------------
// tau vector per batch: TAU[b][i], i<768 -> sin(ts*f_i), else cos(ts*f_{i-768})
// (timestep is a per-batch scalar broadcast across T, so one row per batch.)
// -----------------------------------------------------------------------------
__global__ __launch_bounds__(256) void tau_fill_kernel(
    const int* __restrict__ timesteps, float* __restrict__ TAU)
{
    const int b = blockIdx.x;
    const float ts = (float)timesteps[b];
    const float c = logf(10000.0f) / 768.0f;   // half = HIDDEN/2 = 768
    for (int i = threadIdx.x; i < HIDDEN; i += 256) {
        const int j = (i < 768) ? i : (i - 768);
        const float f = expf(-c * (float)j);
        const float ang = ts * f;
        TAU[(size_t)b * HIDDEN + i] = (i < 768) ? sinf(ang) : cosf(ang);
    }
}

// -----------------------------------------------------------------------------
// Category-deduped GEMV: TADD[b][n] = sum_k TAU[b][k] * W2[cat][1536+k][n]
// Grid over (N-tile, category). Batches sharing a category are processed in
// groups of four with four accumulators, so each category's W2_bot slice
// (9.4 MB) streams from HBM once. Unused-category workgroups exit without
// touching weights. tau[b][k] operands are block-uniform -> scalar loads.
// -----------------------------------------------------------------------------
__global__ __launch_bounds__(256) void tau_gemv_kernel(
    const float* __restrict__ TAU,      // [BATCH, HIDDEN]
    const float* __restrict__ W2,       // [NEMB, 2*HIDDEN, HIDDEN]
    const int*   __restrict__ cat_ids,
    float*       __restrict__ TADD)     // [BATCH, HIDDEN]
{
    const int cat = blockIdx.y;
    const int n   = blockIdx.x * 256 + threadIdx.x;

    const float* wbase = W2 + (size_t)cat * (2 * HIDDEN) * HIDDEN
                            + (size_t)HIDDEN * HIDDEN + n;   // bottom half, col n

    auto process = [&](int c0, int c1, int c2, int c3, int m) {
        const float* t0 = TAU + (size_t)c0 * HIDDEN;
        const float* t1 = TAU + (size_t)c1 * HIDDEN;
        const float* t2 = TAU + (size_t)c2 * HIDDEN;
        const float* t3 = TAU + (size_t)c3 * HIDDEN;
        float a0 = 0.f, a1 = 0.f, a2 = 0.f, a3 = 0.f;
        #pragma unroll 4
        for (int k = 0; k < HIDDEN; ++k) {
            const float w = wbase[(size_t)k * HIDDEN];
            a0 += t0[k] * w;
            a1 += t1[k] * w;
            a2 += t2[k] * w;
            a3 += t3[k] * w;
        }
        TADD[(size_t)c0 * HIDDEN + n] = a0;
        if (m > 1) TADD[(size_t)c1 * HIDDEN + n] = a1;
        if (m > 2) TADD[(size_t)c2 * HIDDEN + n] = a2;
        if (m > 3) TADD[(size_t)c3 * HIDDEN + n] = a3;
    };

    int g0 = 0, g1 = 0, g2 = 0, g3 = 0, cnt = 0;
    for (int b = 0; b < BATCH; ++b) {
        if (cat_ids[b] != cat) continue;
        if      (cnt == 0) g0 = b;
        else if (cnt == 1) g1 = b;
        else if (cnt == 2) g2 = b;
        else               g3 = b;
        if (++cnt == 4) { process(g0, g1, g2, g3, 4); cnt = 0; }
    }
    if (cnt > 0)
        process(g0, (cnt > 1) ? g1 : g0, (cnt > 2) ? g2 : g0, g0, cnt);
}

// -----------------------------------------------------------------------------
// Category-gathered GEMM via V_WMMA_F32_16X16X4_F32.
//   Out[b] (32 x 1536) = act( A[b] (32 x K) @ W[cat_b] (K x 1536)
//                             + bias[cat_b] + row_add[b] )
// Grid: (12 N-tiles of 128 cols, 32 categories). Block: 128 thr = 4 waves.
// Each wave owns 32 output columns (two 16-col B slices) and both 16-row
// M-tiles of one batch -> 4 accumulators; per 4-k step: 4 B b32 + 2 A b64
// loads feed 4 WMMAs (1.5 VMEM/WMMA). Workgroups whose category is unused by
// any batch exit without touching weights (cat dedup: weights stream once).
//
// fp32 WMMA 16x16x4 operand layout:
//   A (16x4): lanes 0-15 = rows with K{0,1}; lanes 16-31 = K{2,3}  -> b64/lane
//   B (4x16): lanes 0-15 = cols with K{0,1}; lanes 16-31 = K{2,3}  -> 2 b32/lane
// -----------------------------------------------------------------------------
template<bool SWISH>
__global__ __launch_bounds__(128) void cat_gemm_f32_wmma(
    const float* __restrict__ A,        // [BATCH*TSEQ, lda]
    const float* __restrict__ W,        // per-cat stride = cat_stride elements
    const float* __restrict__ bias,     // [NEMB, HIDDEN]
    const float* __restrict__ row_add,  // [BATCH, HIDDEN] or nullptr
    float*       __restrict__ Out,      // [BATCH*TSEQ, ldo]
    const int*   __restrict__ cat_ids,  // [BATCH]
    size_t cat_stride, int K, int lda, int ldo)
{
    const int cat    = blockIdx.y;
    const int wave   = threadIdx.x >> 5;
    const int lane   = threadIdx.x & 31;
    const int r16    = lane & 15;
    const int hiHalf = lane >> 4;            // 0: K rows {0,1}, 1: K rows {2,3}
    const int krow   = hiHalf << 1;
    const int n      = blockIdx.x * 128 + wave * 32 + r16;  // slice0 col; slice1 = n+16

    const float* Wc = W + (size_t)cat * cat_stride + (size_t)krow * HIDDEN + n;
    const float  bb0 = bias[cat * HIDDEN + n];
    const float  bb1 = bias[cat * HIDDEN + n + 16];

    for (int b = 0; b < BATCH; ++b) {
        if (cat_ids[b] != cat) continue;     // dedup: only owning WG reads weights

        const float* Ab0 = A + ((size_t)b * TSEQ + r16) * lda + krow;
        const float* Ab1 = Ab0 + (size_t)16 * lda;
        const float* wp  = Wc;

        v8f acc00 = {};   // M-tile 0 (rows 0..15),  cols n..n+15
        v8f acc01 = {};   // M-tile 0,               cols n+16..n+31
        v8f acc10 = {};   // M-tile 1 (rows 16..31), cols n..n+15
        v8f acc11 = {};   // M-tile 1,               cols n+16..n+31

        #pragma unroll 4
        for (int k = 0; k < K; k += 4) {
            v2f bv0, bv1;
            bv0.x = wp[0];            // W[k+krow  ][n]
            bv0.y = wp[HIDDEN];       // W[k+krow+1][n]
            bv1.x = wp[16];           // W[k+krow  ][n+16]
            bv1.y = wp[HIDDEN + 16];  // W[k+krow+1][n+16]
            v2f a0 = *(const v2f*)(Ab0 + k);
            v2f a1 = *(const v2f*)(Ab1 + k);
            acc00 = __builtin_amdgcn_wmma_f32_16x16x4_f32(
                false, a0, false, bv0, (short)0, acc00, false, false);
            acc01 = __builtin_amdgcn_wmma_f32_16x16x4_f32(
                false, a0, false, bv1, (short)0, acc01, false, false);
            acc10 = __builtin_amdgcn_wmma_f32_16x16x4_f32(
                false, a1, false, bv0, (short)0, acc10, false, false);
            acc11 = __builtin_amdgcn_wmma_f32_16x16x4_f32(
                false, a1, false, bv1, (short)0, acc11, false, false);
            wp += (size_t)4 * HIDDEN;
        }

        float add0 = bb0, add1 = bb1;
        if (row_add) {
            add0 += row_add[(size_t)b * HIDDEN + n];
            add1 += row_add[(size_t)b * HIDDEN + n + 16];
        }

        // C/D layout: c[r] holds D[row=r (+8 if hiHalf)][col=r16]
        float* Ob = Out + (size_t)b * TSEQ * ldo + n;
        #pragma unroll
        for (int r = 0; r < 8; ++r) {
            const int row = hiHalf ? (8 + r) : r;
            float v00 = acc00[r] + add0;
            float v01 = acc01[r] + add1;
            float v10 = acc10[r] + add0;
            float v11 = acc11[r] + add1;
            if (SWISH) {
                v00 = v00 / (1.0f + __expf(-v00));
                v01 = v01 / (1.0f + __expf(-v01));
                v10 = v10 / (1.0f + __expf(-v10));
                v11 = v11 / (1.0f + __expf(-v11));
            }
            Ob[(size_t)row * ldo]             = v00;
            Ob[(size_t)row * ldo + 16]        = v01;
            Ob[(size_t)(16 + row) * ldo]      = v10;
            Ob[(size_t)(16 + row) * ldo + 16] = v11;
        }
    }
}

// -----------------------------------------------------------------------------
// Launch: tau fill + deduped tau GEMV (bottom half of W2) + three gathered GEMMs.
// Workspace: Y1 [1024,1536] (6.3MB) | Y2 [1024,1536] (6.3MB)
//            | TAU [32,1536] | TADD [32,1536]  (~13 MB total)
// -----------------------------------------------------------------------------
extern "C" void kernel_launch(void* const* d_in, const int* in_sizes, int n_in,
                              void* d_out, int out_size, void* d_ws, size_t ws_size,
                              hipStream_t stream)
{
    const float* actions   = (const float*)d_in[0];
    const int*   timesteps = (const int*)  d_in[1];
    const int*   cat_ids   = (const int*)  d_in[2];
    const float* W1        = (const float*)d_in[3];
    const float* b1        = (const float*)d_in[4];
    const float* W2        = (const float*)d_in[5];
    const float* b2        = (const float*)d_in[6];
    const float* W3        = (const float*)d_in[7];
    const float* b3        = (const float*)d_in[8];
    float*       out       = (float*)d_out;

    float* Y1   = (float*)d_ws;                             // [1024, 1536]
    float* Y2   = Y1 + (size_t)BATCH * TSEQ * HIDDEN;       // [1024, 1536]
    float* TAU  = Y2 + (size_t)BATCH * TSEQ * HIDDEN;       // [32, 1536]
    float* TADD = TAU + (size_t)BATCH * HIDDEN;             // [32, 1536]

    const dim3 gblk(128);
    const dim3 ggrd(HIDDEN / 128, NEMB);                    // 12 x 32

    // tau row per batch
    tau_fill_kernel<<<dim3(BATCH), dim3(256), 0, stream>>>(timesteps, TAU);

    // Layer 1: actions (32x16) @ W1[c] -> Y1
    cat_gemm_f32_wmma<false><<<ggrd, gblk, 0, stream>>>(
        actions, W1, b1, nullptr, Y1, cat_ids,
        (size_t)ADIM * HIDDEN, ADIM, ADIM, HIDDEN);

    // tau @ W2_bot[c] per batch (deduped by category) -> TADD
    tau_gemv_kernel<<<dim3(HIDDEN / 256, NEMB), dim3(256), 0, stream>>>(
        TAU, W2, cat_ids, TADD);

    // Layer 2: Y1 (32x1536) @ W2_top[c] + TADD[b] + b2 -> swish -> Y2
    cat_gemm_f32_wmma<true><<<ggrd, gblk, 0, stream>>>(
        Y1, W2, b2, TADD, Y2, cat_ids,
        (size_t)(2 * HIDDEN) * HIDDEN, HIDDEN, HIDDEN, HIDDEN);

    // Layer 3: Y2 (32x1536) @ W3[c] + b3 -> out
    cat_gemm_f32_wmma<false><<<ggrd, gblk, 0, stream>>>(
        Y2, W3, b3, nullptr, out, cat_ids,
        (size_t)HIDDEN * HIDDEN, HIDDEN, HIDDEN, HIDDEN);
}